// VectorQuantizerEMA_10307921510592
// MI455X (gfx1250) — compile-verified
//
#include <hip/hip_runtime.h>

#define N_CODES 1024
#define CODE_DIM 256
#define DECAYF 0.99f
#define OMD 0.01f
#define EPSV 1e-5f
#define BATCH 32
#define HW 1024            // 32*32
#define N_TOK (BATCH*HW)   // 32768
#define ZSTRIDE 260        // 256 + 4 pad: bank = (4*tok + d) % 64, conflict-free

typedef __attribute__((ext_vector_type(2))) float v2f;
typedef __attribute__((ext_vector_type(8))) float v8f;

// ---------------- kernel 1: ||e||^2 per code (wave per code) ----------------
__global__ void __launch_bounds__(256) enorm_kernel(const float* __restrict__ embed,
                                                    float* __restrict__ enorm) {
    const int wave = threadIdx.x >> 5;
    const int lane = threadIdx.x & 31;
    const int k = blockIdx.x * 8 + wave;
    const float4* row = (const float4*)(embed + (size_t)k * CODE_DIM);
    float s = 0.f;
    float4 v = row[lane];      s += v.x*v.x + v.y*v.y + v.z*v.z + v.w*v.w;
    v = row[lane + 32];        s += v.x*v.x + v.y*v.y + v.z*v.z + v.w*v.w;
    #pragma unroll
    for (int off = 16; off >= 1; off >>= 1) s += __shfl_xor(s, off);
    if (lane == 0) enorm[k] = s;
}

// ---------------- kernel 2: fused distance GEMM (WMMA fp32) + argmin ----------------
// Block: 256 threads (8 waves), 32 tokens. Waves 0-3 -> token rows 0-15,
// waves 4-7 -> rows 16-31; wave (w&3) covers code tiles [16*(w&3), 16*(w&3)+15].
__global__ void __launch_bounds__(256) argmin_kernel(
    const float* __restrict__ z, const float* __restrict__ embed,
    const float* __restrict__ enorm, int* __restrict__ out_idx,
    float* __restrict__ out_idx_f, float* __restrict__ cs) {
    __shared__ float sZ[32 * ZSTRIDE];
    __shared__ float sVal[32][4];
    __shared__ int   sIdx[32][4];

    const int tid = threadIdx.x;
    const int tok0 = blockIdx.x * 32;

    // Stage 32 tokens (transposed view of z) into LDS. tid&31 = token,
    // tid>>5 = d phase -> global reads coalesced along hw, LDS writes conflict-free.
    {
        const int tl = tid & 31;
        const int dd = tid >> 5;
        const int n  = tok0 + tl;
        const int b  = n >> 10;
        const int hw = n & 1023;
        const float* zp = z + (size_t)b * CODE_DIM * HW + hw;
        #pragma unroll
        for (int d = 0; d < CODE_DIM; d += 8)
            sZ[tl * ZSTRIDE + d + dd] = zp[(size_t)(d + dd) * HW];
    }
    __syncthreads();

    const int wave = tid >> 5;
    const int lane = tid & 31;
    const int sub  = wave >> 2;          // token subtile (0 or 1)
    const int wq   = wave & 3;           // code range within subtile
    const int m    = lane & 15;          // row (A) / col (B) within 16-tile
    const int kb   = (lane >> 4) << 1;   // K sub-offset {0,2}

    const float* aBase = &sZ[(sub * 16 + m) * ZSTRIDE + kb];

    float bestVal[8];
    int   bestIdx[8];
    #pragma unroll
    for (int r = 0; r < 8; ++r) { bestVal[r] = 3.0e38f; bestIdx[r] = 0; }

    for (int tt = 0; tt < 16; ++tt) {
        const int col0 = (wq * 16 + tt) * 16;
        const float* bBase = embed + (size_t)(col0 + m) * CODE_DIM + kb;
        v8f c0 = {}; v8f c1 = {};
        #pragma unroll
        for (int d0 = 0; d0 < 128; d0 += 4) {
            v2f a = *(const v2f*)(aBase + d0);
            v2f b = *(const v2f*)(bBase + d0);
            c0 = __builtin_amdgcn_wmma_f32_16x16x4_f32(
                false, a, false, b, (short)0, c0, false, false);
        }
        #pragma unroll
        for (int d0 = 128; d0 < 256; d0 += 4) {
            v2f a = *(const v2f*)(aBase + d0);
            v2f b = *(const v2f*)(bBase + d0);
            c1 = __builtin_amdgcn_wmma_f32_16x16x4_f32(
                false, a, false, b, (short)0, c1, false, false);
        }
        const float en = enorm[col0 + m];
        const int code = col0 + m;
        #pragma unroll
        for (int r = 0; r < 8; ++r) {
            float d = en - 2.0f * (c0[r] + c1[r]);   // ||e||^2 - 2 x.e
            if (d < bestVal[r] || (d == bestVal[r] && code < bestIdx[r])) {
                bestVal[r] = d; bestIdx[r] = code;
            }
        }
    }

    // C layout: VGPR r = row r (lanes 0-15) / row r+8 (lanes 16-31).
    // Reduce each 16-lane half (xor masks < 16 stay inside the half).
    #pragma unroll
    for (int r = 0; r < 8; ++r) {
        float bv = bestVal[r]; int bi = bestIdx[r];
        #pragma unroll
        for (int off = 8; off >= 1; off >>= 1) {
            float ov = __shfl_xor(bv, off);
            int   oi = __shfl_xor(bi, off);
            if (ov < bv || (ov == bv && oi < bi)) { bv = ov; bi = oi; }
        }
        if (m == 0) {
            const int rowLocal = sub * 16 + r + (lane >> 4) * 8;
            sVal[rowLocal][wq] = bv;
            sIdx[rowLocal][wq] = bi;
        }
    }
    __syncthreads();

    if (tid < 32) {
        float bv = sVal[tid][0]; int bi = sIdx[tid][0];
        #pragma unroll
        for (int q = 1; q < 4; ++q) {
            float ov = sVal[tid][q]; int oi = sIdx[tid][q];
            if (ov < bv || (ov == bv && oi < bi)) { bv = ov; bi = oi; }
        }
        const int g = tok0 + tid;
        out_idx[g] = bi;
        out_idx_f[g] = (float)bi;
        atomicAdd(&cs[bi], 1.0f);
    }
}

// ---------------- kernel 3: gather z_q, scatter es, loss partials ----------------
__global__ void __launch_bounds__(256) scatter_kernel(
    const float* __restrict__ z, const float* __restrict__ embed,
    const int* __restrict__ idx, float* __restrict__ zq_out,
    float* __restrict__ es, float* __restrict__ loss) {
    __shared__ float red[256];
    const int t   = blockIdx.x;          // 0..32767 = (b, d, hw_chunk)
    const int hwc = t & 3;
    const int d   = (t >> 2) & 255;
    const int b   = t >> 10;
    const int hw  = hwc * 256 + threadIdx.x;
    const int n   = b * HW + hw;
    const size_t zoff = (size_t)b * CODE_DIM * HW + (size_t)d * HW + hw;
    const int k = idx[n];
    const float e  = embed[(size_t)k * CODE_DIM + d];
    const float zv = z[zoff];
    zq_out[zoff] = e;                                   // z_q_st == z_q numerically
    atomicAdd(&es[(size_t)k * CODE_DIM + d], zv);
    const float diff = zv - e;
    red[threadIdx.x] = diff * diff;
    __syncthreads();
    #pragma unroll
    for (int s = 128; s > 0; s >>= 1) {
        if (threadIdx.x < s) red[threadIdx.x] += red[threadIdx.x + s];
        __syncthreads();
    }
    if (threadIdx.x == 0) atomicAdd(loss, red[0]);
}

// ---------------- kernel 4: EMA cluster sizes + normalizer + loss ----------------
__global__ void __launch_bounds__(1024) ema_stats_kernel(
    const float* __restrict__ cluster_size, const float* __restrict__ cs,
    const float* __restrict__ loss, float* __restrict__ out_new_cs,
    float* __restrict__ out_loss, float* __restrict__ csnorm) {
    __shared__ float red[1024];
    const int k = threadIdx.x;
    const float ncs = cluster_size[k] * DECAYF + OMD * cs[k];
    out_new_cs[k] = ncs;
    red[k] = ncs;
    __syncthreads();
    #pragma unroll
    for (int s = 512; s > 0; s >>= 1) {
        if (k < s) red[k] += red[k + s];
        __syncthreads();
    }
    const float n = red[0];
    csnorm[k] = (ncs + EPSV) / (n + (float)N_CODES * EPSV) * n;
    if (k == 0) out_loss[0] = loss[0] / (float)((size_t)BATCH * CODE_DIM * HW);
}

// ---------------- kernel 5: EMA embed_avg + normalized embed ----------------
__global__ void __launch_bounds__(256) ema_embed_kernel(
    const float* __restrict__ embed_avg, const float* __restrict__ es,
    const float* __restrict__ csnorm, float* __restrict__ out_new_avg,
    float* __restrict__ out_new_embed) {
    const int k = blockIdx.x;
    const int d = threadIdx.x;
    const size_t o = (size_t)k * CODE_DIM + d;
    const float na = embed_avg[o] * DECAYF + OMD * es[o];
    out_new_avg[o] = na;
    out_new_embed[o] = na / csnorm[k];
}

extern "C" void kernel_launch(void* const* d_in, const int* in_sizes, int n_in,
                              void* d_out, int out_size, void* d_ws, size_t ws_size,
                              hipStream_t stream) {
    const float* z            = (const float*)d_in[0];
    const float* embed        = (const float*)d_in[1];
    const float* cluster_size = (const float*)d_in[2];
    const float* embed_avg    = (const float*)d_in[3];

    float* out          = (float*)d_out;
    float* out_zq       = out;                 // 8388608
    float* out_idx_f    = out + 8388608;       // 32768
    float* out_loss     = out + 8421376;       // 1
    float* out_new_emb  = out + 8421377;       // 262144
    float* out_new_cs   = out + 8683521;       // 1024
    float* out_new_avg  = out + 8684545;       // 262144

    char*  ws        = (char*)d_ws;
    int*   ws_idx    = (int*)ws;                         // 131072 B
    float* ws_cs     = (float*)(ws + 131072);            // 4096 B
    float* ws_es     = (float*)(ws + 135168);            // 1048576 B
    float* ws_loss   = (float*)(ws + 1183744);           // 4 B
    float* ws_csnorm = (float*)(ws + 1183752);           // 4096 B
    float* ws_enorm  = (float*)(ws + 1187848);           // 4096 B

    // zero cs + es + loss (contiguous) each call; graph-capturable
    hipMemsetAsync(ws + 131072, 0, 1052676, stream);

    enorm_kernel<<<N_CODES / 8, 256, 0, stream>>>(embed, ws_enorm);
    argmin_kernel<<<N_TOK / 32, 256, 0, stream>>>(z, embed, ws_enorm,
                                                  ws_idx, out_idx_f, ws_cs);
    scatter_kernel<<<32768, 256, 0, stream>>>(z, embed, ws_idx,
                                              out_zq, ws_es, ws_loss);
    ema_stats_kernel<<<1, 1024, 0, stream>>>(cluster_size, ws_cs, ws_loss,
                                             out_new_cs, out_loss, ws_csnorm);
    ema_embed_kernel<<<N_CODES, 256, 0, stream>>>(embed_avg, ws_es, ws_csnorm,
                                                  out_new_avg, out_new_emb);
}